// TitanFusion_18597208392430
// MI455X (gfx1250) — compile-verified
//
#include <hip/hip_runtime.h>
#include <stddef.h>

// ---- precision / shape constants from the reference ----
#define Dk   1024   // feature dim (K of gating GEMM)
#define NB   4      // number of branches (N of gating GEMM)
#define ROWS_PER_BLOCK 64
#define WAVES_PER_BLOCK 4          // wave32 -> 128 threads
#define THREADS (WAVES_PER_BLOCK * 32)

typedef float v2f __attribute__((ext_vector_type(2)));
typedef float v4f __attribute__((ext_vector_type(4)));
typedef float v8f __attribute__((ext_vector_type(8)));

__global__ __launch_bounds__(THREADS)
void moe_gate_fuse_kernel(const float* __restrict__ x,
                          const float* __restrict__ branches,
                          const float* __restrict__ gate_w,
                          const float* __restrict__ gate_b,
                          float* __restrict__ out)
{
    __shared__ float gwLds[Dk * NB];                 // 16 KB gate weights
    __shared__ float gbLds[NB];
    __shared__ float logitsLds[ROWS_PER_BLOCK * NB]; // 1 KB
    __shared__ float wLds[ROWS_PER_BLOCK * NB];      // 1 KB softmax weights

    const int tid  = threadIdx.x;
    const int wave = tid >> 5;
    const int lane = tid & 31;
    const int rowBlockBase = blockIdx.x * ROWS_PER_BLOCK;

    // ---------- stage gate_w / gate_b into LDS (coalesced b128 loads) ----------
    for (int i = tid; i < (Dk * NB) / 4; i += THREADS) {
        ((v4f*)gwLds)[i] = ((const v4f*)gate_w)[i];
    }
    if (tid < NB) gbLds[tid] = gate_b[tid];
    __syncthreads();

    // ---------- phase 1: gating logits via V_WMMA_F32_16X16X4_F32 ----------
    // Each wave owns a 16-row tile: rows [rowBlockBase + wave*16, +16).
    // A (16x4 f32) per-lane layout: lanes 0-15 -> M=lane, VGPR {K,K+1};
    //                               lanes 16-31 -> M=lane-16, VGPR {K+2,K+3}.
    // B (4x16 f32): VGPR v, lanes 0-15 -> K=v+0..1? half split: lane>=16 -> K+2.
    // Only N<4 columns are real gate_w; others are zero-masked.
    {
        const int  m     = lane & 15;          // tile row
        const int  kh    = (lane >> 4) * 2;    // K sub-offset of this half-wave
        const int  n     = lane & 15;          // B/C column
        const float nMask = (n < NB) ? 1.0f : 0.0f;

        const float* aPtr = x + (size_t)(rowBlockBase + wave * 16 + m) * Dk + kh;
        const float* bPtr = gwLds + kh * NB + (n & (NB - 1));

        v8f acc = {0.f, 0.f, 0.f, 0.f, 0.f, 0.f, 0.f, 0.f};
        for (int k0 = 0; k0 < Dk; k0 += 4) {
            // streaming read of x: non-temporal (one-pass data)
            v2f a = __builtin_nontemporal_load((const v2f*)(aPtr + k0));
            v2f b;
            b.x = bPtr[k0 * NB]        * nMask;   // B[k0+kh+0][n]
            b.y = bPtr[k0 * NB + NB]   * nMask;   // B[k0+kh+1][n]
            acc = __builtin_amdgcn_wmma_f32_16x16x4_f32(
                      /*neg_a=*/false, a, /*neg_b=*/false, b,
                      /*c_mod=*/(short)0, acc,
                      /*reuse_a=*/false, /*reuse_b=*/false);
        }

        // Scatter valid logits (N<4) to LDS.
        // C layout: acc[v] @ lane L = element (M = v + 8*(L>>4), N = L&15).
        if ((lane & 12) == 0) {               // lanes 0-3 and 16-19
            const int n4   = lane & 3;
            const int mOff = (lane >> 4) * 8;
#pragma unroll
            for (int v = 0; v < 8; ++v) {
                logitsLds[(wave * 16 + mOff + v) * NB + n4] = acc[v];
            }
        }
    }
    __syncthreads();

    // ---------- phase 2: per-row top-2 mask + softmax ----------
    if (tid < ROWS_PER_BLOCK) {
        float l[NB];
#pragma unroll
        for (int n = 0; n < NB; ++n) l[n] = logitsLds[tid * NB + n] + gbLds[n];

        // kth = 2nd-largest (duplicates count); reference keeps logits >= kth
        float m1 = -__builtin_inff(), m2 = -__builtin_inff();
#pragma unroll
        for (int n = 0; n < NB; ++n) {
            float v = l[n];
            if (v > m1)      { m2 = m1; m1 = v; }
            else if (v > m2) { m2 = v; }
        }

        float s = 0.f, w[NB];
#pragma unroll
        for (int n = 0; n < NB; ++n) {
            float e = (l[n] >= m2) ? __expf(l[n] - m1) : 0.f;
            w[n] = e;
            s += e;
        }
        const float inv = 1.0f / s;
#pragma unroll
        for (int n = 0; n < NB; ++n) wLds[tid * NB + n] = w[n] * inv;
    }
    __syncthreads();

    // ---------- phase 3: fused = sum_n w[n] * branches[n] (HBM-bound stream) ----------
    // 64 rows * 256 float4 per row; consecutive lanes -> consecutive d (coalesced).
    const int totalF4 = ROWS_PER_BLOCK * (Dk / 4);
    for (int i = tid; i < totalF4; i += THREADS) {
        const int r  = i >> 8;        // Dk/4 == 256
        const int d4 = i & 255;
        const size_t row = (size_t)(rowBlockBase + r);

        const float w0 = wLds[r * NB + 0];
        const float w1 = wLds[r * NB + 1];
        const float w2 = wLds[r * NB + 2];
        const float w3 = wLds[r * NB + 3];

        const v4f* bbase = (const v4f*)(branches + row * (size_t)NB * Dk) + d4;
        v4f b0 = __builtin_nontemporal_load(bbase);
        v4f b1 = __builtin_nontemporal_load(bbase + (Dk / 4));
        v4f b2 = __builtin_nontemporal_load(bbase + 2 * (Dk / 4));
        v4f b3 = __builtin_nontemporal_load(bbase + 3 * (Dk / 4));

        v4f o;
        o.x = __builtin_fmaf(w0, b0.x, __builtin_fmaf(w1, b1.x, __builtin_fmaf(w2, b2.x, w3 * b3.x)));
        o.y = __builtin_fmaf(w0, b0.y, __builtin_fmaf(w1, b1.y, __builtin_fmaf(w2, b2.y, w3 * b3.y)));
        o.z = __builtin_fmaf(w0, b0.z, __builtin_fmaf(w1, b1.z, __builtin_fmaf(w2, b2.z, w3 * b3.z)));
        o.w = __builtin_fmaf(w0, b0.w, __builtin_fmaf(w1, b1.w, __builtin_fmaf(w2, b2.w, w3 * b3.w)));

        __builtin_nontemporal_store(o, (v4f*)(out + row * Dk) + d4);
    }
}

extern "C" void kernel_launch(void* const* d_in, const int* in_sizes, int n_in,
                              void* d_out, int out_size, void* d_ws, size_t ws_size,
                              hipStream_t stream) {
    const float* x        = (const float*)d_in[0];   // (B, L, D)
    const float* branches = (const float*)d_in[1];   // (B, L, NB, D)
    const float* gate_w   = (const float*)d_in[2];   // (D, NB)
    const float* gate_b   = (const float*)d_in[3];   // (NB,)
    float*       out      = (float*)d_out;           // (B, L, D)

    const int rows   = in_sizes[0] / Dk;             // B*L = 16384
    const int blocks = rows / ROWS_PER_BLOCK;        // 256

    hipLaunchKernelGGL(moe_gate_fuse_kernel,
                       dim3(blocks), dim3(THREADS), 0, stream,
                       x, branches, gate_w, gate_b, out);
}